// DetectionLoss_81329500717556
// MI455X (gfx1250) — compile-verified
//
#include <hip/hip_runtime.h>

// Detection loss (focal cls + DFL box) for B=16, N=50, C=144 (=4*16 bins + 80 classes),
// layers 80x80 / 40x40 / 20x20. Output: (total, total_cls, total_box) as 3 f32.
//
// Kernel 1: one wave32 per (b,n,layer) entry (2400 waves). Wave-shuffle softmax
//           reductions; per-block partial (cls,box) pair -> d_ws (deterministic).
// Kernel 2: single wave reduces 300 pairs with chained V_WMMA_F32_16X16X4_F32
//           (A = data tile, B = ones => row-sum reduction; row parity separates
//           cls from box since pairs are interleaved in memory).

#define NTGT   50
#define NBLK1  300   // 2400 entries / 8 waves per block
#define NPAIRS 300

typedef __attribute__((ext_vector_type(2))) float v2f;
typedef __attribute__((ext_vector_type(8))) float v8f;

__global__ __launch_bounds__(256) void det_loss_main(
    const float* __restrict__ f0, const float* __restrict__ f1,
    const float* __restrict__ f2, const float* __restrict__ tgt,
    float* __restrict__ partials)
{
    __shared__ float s_cls[8];
    __shared__ float s_box[8];

    const int lane = threadIdx.x & 31;
    const int wave = threadIdx.x >> 5;
    const int g    = blockIdx.x * 8 + wave;      // 0..2399
    const int layer = g / 800;                   // 0,1,2
    const int r     = g - layer * 800;
    const int b     = r / NTGT;
    const int n     = r - b * NTGT;

    const int W  = 80 >> layer;                  // 80,40,20 (H==W)
    const int HW = W * W;
    const float* feat = (layer == 0) ? f0 : (layer == 1 ? f1 : f2);

    const float* t = tgt + ((size_t)b * NTGT + n) * 6;
    const int   cls_t = (int)t[0];
    const float cx = t[1], cy = t[2], w = t[3], h = t[4];
    const int   t5 = (int)t[5];

    float clsL = 0.0f, boxL = 0.0f;
    if (t5 == layer) {   // wave-uniform branch
        int fx = (int)(cx * (float)W); fx = min(max(fx, 0), W - 1);
        int fy = (int)(cy * (float)W); fy = min(max(fy, 0), W - 1);
        const float* base = feat + (size_t)b * 144 * HW + (size_t)fy * W + fx;

        // ---------- focal classification loss over 80 classes ----------
        // lane covers classes lane, lane+32, and lane+64 (lane<16 only)
        float l0 = base[(size_t)(64 + lane) * HW];
        float l1 = base[(size_t)(64 + lane + 32) * HW];
        float l2 = (lane < 16) ? base[(size_t)(64 + lane + 64) * HW] : -3.0e38f;
        float m = fmaxf(fmaxf(l0, l1), l2);
        #pragma unroll
        for (int off = 16; off >= 1; off >>= 1)
            m = fmaxf(m, __shfl_xor(m, off, 32));
        float se = expf(l0 - m) + expf(l1 - m) + ((lane < 16) ? expf(l2 - m) : 0.0f);
        #pragma unroll
        for (int off = 16; off >= 1; off >>= 1)
            se += __shfl_xor(se, off, 32);
        float lt = base[(size_t)(64 + cls_t) * HW];
        float ce = -(lt - m - logf(se));          // -log_softmax[cls_t]
        float pt = expf(-ce);
        float om = 1.0f - pt;
        clsL = om * om * ce;                      // ALPHA=1, GAMMA=2

        // ---------- DFL box loss: 4 sides x 16 bins ----------
        // two passes; each pass: lanes 0-15 do side 2p, lanes 16-31 do side 2p+1
        const int half = lane >> 4;
        const int bin  = lane & 15;
        float boxAcc = 0.0f;
        #pragma unroll
        for (int p = 0; p < 2; ++p) {
            int side = p * 2 + half;
            float v = base[(size_t)(side * 16 + bin) * HW];
            float gm = v;
            #pragma unroll
            for (int off = 8; off >= 1; off >>= 1)
                gm = fmaxf(gm, __shfl_xor(gm, off, 16));
            float gs = expf(v - gm);
            #pragma unroll
            for (int off = 8; off >= 1; off >>= 1)
                gs += __shfl_xor(gs, off, 16);
            float lsm = gm + logf(gs);
            float lp = v - lsm;                   // log_softmax of this side
            // sides 0,2 target w/2*W ; sides 1,3 target h/2*H (H==W)
            float gt = ((side & 1) == 0) ? (w * 0.5f * (float)W)
                                         : (h * 0.5f * (float)W);
            float tg = fminf(fmaxf(gt, 0.0f), 15.0f - 1e-6f);
            int   li = (int)floorf(tg);
            float wl = (float)(li + 1) - tg;
            float wr = tg - (float)li;
            float lpl = __shfl(lp, li, 16);
            float lpr = __shfl(lp, min(li + 1, 15), 16);
            boxAcc += -(lpl * wl + lpr * wr);
        }
        // combine the two 16-lane halves -> all 4 sides
        boxL = boxAcc + __shfl_xor(boxAcc, 16, 32);
    }

    if (lane == 0) { s_cls[wave] = clsL; s_box[wave] = boxL; }
    __syncthreads();
    if (threadIdx.x == 0) {
        float c = 0.0f, bx = 0.0f;
        #pragma unroll
        for (int i = 0; i < 8; ++i) { c += s_cls[i]; bx += s_box[i]; }
        partials[blockIdx.x * 2 + 0] = c;
        partials[blockIdx.x * 2 + 1] = bx;
    }
}

// Single-wave WMMA matmul-reduction of NPAIRS interleaved (cls,box) pairs.
// A[m][k] = partials[tile*64 + k*16 + m], B = ones(4x16):
//   D[m][n] = sum_k A[m][k]  (replicated over n). Interleaving => even rows m
// accumulate cls partials, odd rows accumulate box partials.
__global__ __launch_bounds__(32) void det_loss_reduce(
    const float* __restrict__ partials, float* __restrict__ out)
{
    const int lane = threadIdx.x & 31;
    const int m  = lane & 15;
    const int k0 = (lane >> 4) * 2;              // A layout: lanes 0-15 K=0,1; 16-31 K=2,3
    const int nfl = NPAIRS * 2;                  // 600 floats
    const int ntiles = (nfl + 63) / 64;          // 10 tiles

    v2f ones; ones.x = 1.0f; ones.y = 1.0f;      // every element of B is 1
    v8f acc = {};

    for (int tile = 0; tile < ntiles; ++tile) {
        int i0 = tile * 64 + k0 * 16 + m;
        int i1 = i0 + 16;
        v2f a;
        a.x = (i0 < nfl) ? partials[i0] : 0.0f;
        a.y = (i1 < nfl) ? partials[i1] : 0.0f;
        // (neg_a, A, neg_b, B, c_mod, C, reuse_a, reuse_b); NEG must be 0 for f32
        acc = __builtin_amdgcn_wmma_f32_16x16x4_f32(
            false, a, false, ones, (short)0, acc, false, false);
    }

    // C/D layout: VGPR j = row j (lanes 0-15) / row j+8 (lanes 16-31)
    float ev = acc[0] + acc[2] + acc[4] + acc[6];   // even rows -> cls
    float od = acc[1] + acc[3] + acc[5] + acc[7];   // odd rows  -> box
    ev += __shfl_xor(ev, 16, 32);                   // rows 0-7 + rows 8-15
    od += __shfl_xor(od, 16, 32);

    if (lane == 0) {
        out[0] = ev + od;   // total (CLS_WEIGHT = BOX_WEIGHT = 1)
        out[1] = ev;        // total_cls
        out[2] = od;        // total_box
    }
}

extern "C" void kernel_launch(void* const* d_in, const int* in_sizes, int n_in,
                              void* d_out, int out_size, void* d_ws, size_t ws_size,
                              hipStream_t stream) {
    const float* f0 = (const float*)d_in[0];
    const float* f1 = (const float*)d_in[1];
    const float* f2 = (const float*)d_in[2];
    const float* tg = (const float*)d_in[3];
    float* ws  = (float*)d_ws;
    float* out = (float*)d_out;

    det_loss_main<<<NBLK1, 256, 0, stream>>>(f0, f1, f2, tg, ws);
    det_loss_reduce<<<1, 32, 0, stream>>>(ws, out);
}